// RelGraphConvLayer_41489384079586
// MI455X (gfx1250) — compile-verified
//
#include <hip/hip_runtime.h>
#include <hip/hip_bf16.h>
#include <math.h>
#include <stdint.h>

#define DDIM 128
#define NNODES 100000
#define NRELS 4
#define NEDGES 250000
#define A_STRIDE 132   // padded LDS row stride (floats) to avoid bank conflicts

typedef __attribute__((ext_vector_type(2))) float v2f;
typedef __attribute__((ext_vector_type(8))) float v8f;

// ---------------------------------------------------------------------------
// Zero a range of floats
// ---------------------------------------------------------------------------
__global__ void rgcl_zero_f32(float* __restrict__ p, size_t n) {
    size_t i = (size_t)blockIdx.x * blockDim.x + threadIdx.x;
    if (i < n) p[i] = 0.0f;
}

// ---------------------------------------------------------------------------
// Edge scatter: 32 lanes per edge, each lane handles 4 contiguous dims.
// msg[dst] += x[src];  deg[dst] += 1
// ---------------------------------------------------------------------------
__global__ __launch_bounds__(256)
void rgcl_scatter(const float* __restrict__ x,
                  const int* __restrict__ src,
                  const int* __restrict__ dst,
                  float* __restrict__ msg,
                  float* __restrict__ deg,
                  int nEdges) {
    int idx  = blockIdx.x * blockDim.x + threadIdx.x;
    int e    = idx >> 5;
    int lane = idx & 31;
    if (e >= nEdges) return;
    int s = src[e];
    int d = dst[e];
    const float4* xs = (const float4*)(x + (size_t)s * DDIM);
    float4 v = xs[lane];                       // 32 lanes * 4 floats = 128
    float* md = msg + (size_t)d * DDIM + lane * 4;
    atomicAdd(md + 0, v.x);
    atomicAdd(md + 1, v.y);
    atomicAdd(md + 2, v.z);
    atomicAdd(md + 3, v.w);
    if (lane == 0) atomicAdd(deg + d, 1.0f);
}

// ---------------------------------------------------------------------------
// Async copy of one 16x128 f32 tile (row-major, contiguous rows of D floats)
// into padded LDS (A_STRIDE floats per row), via GLOBAL_LOAD_ASYNC_TO_LDS_B128.
// Each of the 256 threads covers 32 bytes (two B128 transfers, offset:0 / 16).
// INST_OFFSET is added to BOTH the global and LDS addresses (ISA 15.18.3).
// ---------------------------------------------------------------------------
__device__ __forceinline__
void async_copy_tile(const float* __restrict__ gsrc, float* lds_tile,
                     int rowBase, int tid) {
    int r  = tid >> 4;               // 0..15
    int c0 = (tid & 15) * 8;         // 0..120
    const float* gp = gsrc + (size_t)(rowBase + r) * DDIM + c0;
    uint32_t l0 = (uint32_t)(uintptr_t)&lds_tile[r * A_STRIDE + c0];
    uint64_t g0 = (uint64_t)(uintptr_t)gp;
    asm volatile("global_load_async_to_lds_b128 %0, %1, off"
                 :: "v"(l0), "v"(g0) : "memory");
    asm volatile("global_load_async_to_lds_b128 %0, %1, off offset:16"
                 :: "v"(l0), "v"(g0) : "memory");
}

__device__ __forceinline__ void wait_async_lds() {
    asm volatile("s_wait_asynccnt 0x0" ::: "memory");
}

// ---------------------------------------------------------------------------
// Per-relation GEMM:  hsum += relu( inv(deg) * (msg @ W) + bias )
// (normalization folded into the epilogue: diag(inv).msg.W == diag(inv).(msg.W))
// Block = 256 threads = 8 waves; block computes 16 rows x 128 cols.
// Each wave owns one 16x16 tile, K stepped by 4 via v_wmma_f32_16x16x4_f32.
// ---------------------------------------------------------------------------
__global__ __launch_bounds__(256)
void rgcl_rel_gemm(const float* __restrict__ msg,
                   const float* __restrict__ deg,
                   const float* __restrict__ W,      // [D][D] row-major, W[k][n]
                   const float* __restrict__ bias,   // [D]
                   float* __restrict__ hsum) {
    __shared__ float As[16 * A_STRIDE];
    __shared__ float invDeg[16];

    const int rowBase = blockIdx.x * 16;
    const int tid  = threadIdx.x;
    const int wave = tid >> 5;
    const int lane = tid & 31;

    async_copy_tile(msg, As, rowBase, tid);
    if (tid < 16) invDeg[tid] = 1.0f / fmaxf(deg[rowBase + tid], 1.0f);
    wait_async_lds();
    __syncthreads();

    const int colBase = wave * 16;
    const int mn = lane & 15;            // A row index == B col index for this lane
    const int kh = (lane >> 4) * 2;      // lanes 16..31 carry K+2, K+3

    v8f c = {};
    for (int kk = 0; kk < DDIM; kk += 4) {
        v2f a, b;
        const float* ap = &As[mn * A_STRIDE + kk + kh];
        a.x = ap[0];
        a.y = ap[1];
        const float* wp = W + (size_t)(kk + kh) * DDIM + colBase + mn;
        b.x = wp[0];
        b.y = wp[DDIM];
        c = __builtin_amdgcn_wmma_f32_16x16x4_f32(
                false, a, false, b, (short)0, c, false, false);
    }

    // Epilogue: per-row degree scale + bias + relu, accumulate into hsum
    const int n     = lane & 15;
    const int mBase = (lane >> 4) * 8;
    float bval = bias[colBase + n];
    float* hp = hsum + (size_t)(rowBase + mBase) * DDIM + colBase + n;
    #pragma unroll
    for (int i = 0; i < 8; ++i) {
        float h = fmaxf(c[i] * invDeg[mBase + i] + bval, 0.0f);
        hp[(size_t)i * DDIM] += h;
    }
}

// ---------------------------------------------------------------------------
// Finalize:  out = relu( silu(hsum/4) + x @ loopW + node_bias )
// ---------------------------------------------------------------------------
__global__ __launch_bounds__(256)
void rgcl_final_gemm(const float* __restrict__ x,
                     const float* __restrict__ loopW,   // [D][D]
                     const float* __restrict__ nbias,   // [D]
                     const float* __restrict__ hsum,
                     float* __restrict__ out) {
    __shared__ float As[16 * A_STRIDE];

    const int rowBase = blockIdx.x * 16;
    const int tid  = threadIdx.x;
    const int wave = tid >> 5;
    const int lane = tid & 31;

    async_copy_tile(x, As, rowBase, tid);
    wait_async_lds();
    __syncthreads();

    const int colBase = wave * 16;
    const int mn = lane & 15;
    const int kh = (lane >> 4) * 2;

    v8f c = {};
    for (int kk = 0; kk < DDIM; kk += 4) {
        v2f a, b;
        const float* ap = &As[mn * A_STRIDE + kk + kh];
        a.x = ap[0];
        a.y = ap[1];
        const float* wp = loopW + (size_t)(kk + kh) * DDIM + colBase + mn;
        b.x = wp[0];
        b.y = wp[DDIM];
        c = __builtin_amdgcn_wmma_f32_16x16x4_f32(
                false, a, false, b, (short)0, c, false, false);
    }

    const int n     = lane & 15;
    const int mBase = (lane >> 4) * 8;
    float bval = nbias[colBase + n];
    const float* hp = hsum + (size_t)(rowBase + mBase) * DDIM + colBase + n;
    float* op = out + (size_t)(rowBase + mBase) * DDIM + colBase + n;
    #pragma unroll
    for (int i = 0; i < 8; ++i) {
        float hs  = hp[(size_t)i * DDIM] * 0.25f;           // mean over 4 relations
        float sil = hs / (1.0f + __expf(-hs));              // SiLU
        float o   = sil + c[i] + bval;
        op[(size_t)i * DDIM] = fmaxf(o, 0.0f);
    }
}

// ---------------------------------------------------------------------------
// Host-side orchestration (graph-capture safe: only stream launches)
// ---------------------------------------------------------------------------
extern "C" void kernel_launch(void* const* d_in, const int* in_sizes, int n_in,
                              void* d_out, int out_size, void* d_ws, size_t ws_size,
                              hipStream_t stream) {
    const float* x          = (const float*)d_in[0];
    const float* weight     = (const float*)d_in[1];   // [R][D][D]
    const float* conv_bias  = (const float*)d_in[2];   // [R][D]
    const float* loop_weight= (const float*)d_in[3];   // [D][D]
    const float* node_bias  = (const float*)d_in[4];   // [1][D]
    const int*   src        = (const int*)d_in[5];     // [R][E]
    const int*   dst        = (const int*)d_in[6];     // [R][E]
    float* out = (float*)d_out;

    const int N = NNODES, R = NRELS, E = NEDGES;
    const size_t ND = (size_t)N * DDIM;

    // ws layout: msg[N*D] | deg[N] | hsum[N*D]  (msg+deg contiguous for one zero pass)
    float* msg  = (float*)d_ws;
    float* deg  = msg + ND;
    float* hsum = deg + N;

    // zero hsum once
    {
        size_t n = ND;
        int blocks = (int)((n + 255) / 256);
        rgcl_zero_f32<<<blocks, 256, 0, stream>>>(hsum, n);
    }

    const int gemmBlocks = N / 16;                 // 100000/16 = 6250 exact
    const int scatterBlocks = (E * 32 + 255) / 256;

    for (int r = 0; r < R; ++r) {
        // zero msg + deg (contiguous)
        {
            size_t n = ND + (size_t)N;
            int blocks = (int)((n + 255) / 256);
            rgcl_zero_f32<<<blocks, 256, 0, stream>>>(msg, n);
        }
        rgcl_scatter<<<scatterBlocks, 256, 0, stream>>>(
            x, src + (size_t)r * E, dst + (size_t)r * E, msg, deg, E);
        rgcl_rel_gemm<<<gemmBlocks, 256, 0, stream>>>(
            msg, deg,
            weight + (size_t)r * DDIM * DDIM,
            conv_bias + (size_t)r * DDIM,
            hsum);
    }

    rgcl_final_gemm<<<gemmBlocks, 256, 0, stream>>>(
        x, loop_weight, node_bias, hsum, out);
}